// HashGridPyTorch_20916490731577
// MI455X (gfx1250) — compile-verified
//
#include <hip/hip_runtime.h>
#include <stdint.h>

// ---------------------------------------------------------------------------
// Multires hash-grid embedding lookup (instant-NGP style, no interpolation).
//   L=16 levels, F=2 feats, base res 16, log2_hashmap=19, B=2^20 points.
// Pure gather: memory-roofline bound (~200 MB @ 23.3 TB/s ~= 8.6 us).
// WMMA is inapplicable (zero matrix FLOPs). CDNA5-specific leverage:
//   * wave32 coalescing (4 threads/point, 32B contiguous per lane on store)
//   * NT-hinted streaming stores so the 128 MB output never evicts the
//     57 MB table working set from the 192 MB L2
//   * global_prefetch_b8 (near scope) on the streaming input
//
// VALU reduction: res_l = 2^(l+4), and x*2^k is exact in f32 (exponent-only
// scaling). Hence floor(x*res_l) == floor(x*2^19) >> (15-l) for all levels,
// with all clamps folding into one clamp of the finest-level coordinate:
//   u = min(u32(max(floor(x*524288), 0)), 524287);  ix_l = u >> (15-l)
// This is bit-exact vs. the reference's per-level float floor/clip.
// ---------------------------------------------------------------------------

typedef float f32x2 __attribute__((ext_vector_type(2)));
typedef float f32x4 __attribute__((ext_vector_type(4)));

#define HG_HASH_MASK 0x7FFFFu

// Per-level (size-1) / offset.
//   sizes:   l0=17^3=4913, l1=33^3=35937, l2=65^3=274625, l>=3: 2^19
//   offsets: 0, 4913, 40850, 315475, then +524288 per level
__device__ __forceinline__ uint32_t hg_index(uint32_t ux, uint32_t uy, uint32_t uz,
                                             uint32_t l) {
    const uint32_t k = 15u - l;
    uint32_t h = (ux >> k) * 73856093u
               + (uy >> k) * 19349663u
               + (uz >> k) * 83492791u;     // uint32 wraparound; 2^19 | 2^32
    h &= HG_HASH_MASK;

    uint32_t szm1, off;
    if (l >= 3u) {
        szm1 = 524287u;                      // min() is a no-op here but cheap
        off  = 315475u + ((l - 3u) << 19);
    } else {
        szm1 = (l == 0u) ? 4912u : ((l == 1u) ? 35936u : 274624u);
        off  = (l == 0u) ? 0u    : ((l == 1u) ? 4913u  : 40850u);
    }
    h = (h < szm1) ? h : szm1;               // min(h, size-1)
    return h + off;
}

// Clamped finest-level (res=2^19) integer coordinate from normalized coord.
__device__ __forceinline__ uint32_t hg_fine_coord(float x01) {
    // x01*524288 is exact (power-of-2 scale); integers <= 2^19 exact in f32.
    float f = floorf(x01 * 524288.0f);
    f = fmaxf(f, 0.0f);
    uint32_t u = (uint32_t)f;
    return (u < 524287u) ? u : 524287u;
}

__global__ __launch_bounds__(256) void hashgrid_gather_kernel(
    const float* __restrict__ inputs,   // [B,3]
    const float* __restrict__ tables,   // [7131219, 2]
    const int*   __restrict__ size_ptr, // scalar
    float*       __restrict__ out,      // [B,32]
    uint32_t B)
{
    const uint32_t tid = blockIdx.x * 256u + threadIdx.x;
    const uint32_t b   = tid >> 2;       // point index (4 threads per point)
    const uint32_t q   = tid & 3u;       // level quartet: levels 4q .. 4q+3
    if (b >= B) return;

    const float s   = (float)(*size_ptr);   // uniform scalar load + cvt
    const float inv = 0.5f / s;             // one IEEE div per thread

    // 4 lanes read the same 12 bytes -> single request, cache broadcast
    const float* p = inputs + (size_t)b * 3u;
    const float x = (p[0] + s) * inv;       // matches reference rounding:
    const float y = (p[1] + s) * inv;       // (p+s)/(2s) then exact *2^19
    const float z = (p[2] + s) * inv;

    // Speculative near-scope prefetch of the input stream (~8K points ahead)
    if (q == 0u) {
        const uint32_t pb = b + 8192u;
        if (pb < B) __builtin_prefetch(inputs + (size_t)pb * 3u, 0, 3);
    }

    const uint32_t ux = hg_fine_coord(x);
    const uint32_t uy = hg_fine_coord(y);
    const uint32_t uz = hg_fine_coord(z);

    const uint32_t l0 = q << 2;
    const uint32_t i0 = hg_index(ux, uy, uz, l0 + 0u);
    const uint32_t i1 = hg_index(ux, uy, uz, l0 + 1u);
    const uint32_t i2 = hg_index(ux, uy, uz, l0 + 2u);
    const uint32_t i3 = hg_index(ux, uy, uz, l0 + 3u);

    // Four independent 8-byte gathers (global_load_b64); the 57 MB table is
    // L2-resident (192 MB L2), so these are L2 hits after first touch.
    const f32x2* __restrict__ t2 = (const f32x2*)tables;
    const f32x2 e0 = t2[i0];
    const f32x2 e1 = t2[i1];
    const f32x2 e2 = t2[i2];
    const f32x2 e3 = t2[i3];

    const f32x4 lo = { e0.x, e0.y, e1.x, e1.y };
    const f32x4 hi = { e2.x, e2.y, e3.x, e3.y };

    // Streaming output: 32 contiguous bytes per lane, non-temporal so the
    // 128 MB output stream doesn't evict the hash tables from L2.
    float* o = out + (size_t)b * 32u + (size_t)(q << 3);
    __builtin_nontemporal_store(lo, (f32x4*)(o));
    __builtin_nontemporal_store(hi, (f32x4*)(o + 4));
}

extern "C" void kernel_launch(void* const* d_in, const int* in_sizes, int n_in,
                              void* d_out, int out_size, void* d_ws, size_t ws_size,
                              hipStream_t stream) {
    (void)n_in; (void)out_size; (void)d_ws; (void)ws_size;
    const float* inputs = (const float*)d_in[0];   // [B,3] f32
    const float* tables = (const float*)d_in[1];   // [7131219,2] f32
    const int*   size_p = (const int*)d_in[2];     // scalar int
    float*       out    = (float*)d_out;           // [B,32] f32

    const uint32_t B     = (uint32_t)(in_sizes[0] / 3);
    const uint32_t total = B * 4u;                 // 4 threads per point
    const uint32_t block = 256u;                   // 8 wave32 per workgroup
    const uint32_t grid  = (total + block - 1u) / block;

    hashgrid_gather_kernel<<<dim3(grid), dim3(block), 0, stream>>>(
        inputs, tables, size_p, out, B);
}